// Projection_39573828666038
// MI455X (gfx1250) — compile-verified
//
#include <hip/hip_runtime.h>

#define NB 16
#define CHN 3
#define HH 384
#define WW 512
#define HW (HH * WW)
#define EPSF 1e-7f

typedef __attribute__((ext_vector_type(2))) float v2f;
typedef __attribute__((ext_vector_type(8))) float v8f;

// ---------------------------------------------------------------------------
// Kernel 1: per-batch A = K @ (R @ K_inv) via two V_WMMA_F32_16X16X4_F32.
// One wave (32 lanes) per batch. 3x3 matrices zero-padded into 16x4 / 4x16
// fragments, packed through LDS "fragment images" so every lane's operand
// fetch is a single branchless ds_load.
//
// f32 A-frag (16x4, MxK), per ISA 7.12.2:
//   VGPR0: lanes 0-15 -> (m=lane, k=0), lanes 16-31 -> (m=lane-16, k=2)
//   VGPR1: lanes 0-15 -> (m=lane, k=1), lanes 16-31 -> (m=lane-16, k=3)
// f32 B-frag (4x16, KxN): same with (k, n=lane) roles.
// f32 D (16x16): VGPR i, lanes 0-15 -> D[i][lane]  (i = 0..7)
//
// LDS image layout: img[k*16 + col] = Mat[row=col][k] (A-frag, transposed)
//                   img[k*16 + col] = Mat[k][n=col]   (B-frag, direct)
// Lane l reads img[koff + m] / img[koff + 16 + m], koff = (l & 16) * 2.
//
// Trig: angles are 0.05*N(0,1) (<< pi), so hardware v_sin/v_cos (__sinf /
// __cosf) are well within tolerance and avoid the libm range-reduction blob.
// ---------------------------------------------------------------------------
__global__ __launch_bounds__(32) void compose_A_kernel(
    const float* __restrict__ pose, const float* __restrict__ Km,
    const float* __restrict__ Ki, float* __restrict__ Aout)
{
    __shared__ float smem[256];
    constexpr int RA = 0;    // A-frag image of R
    constexpr int KI = 64;   // B-frag image of K_inv
    constexpr int KA = 128;  // A-frag image of K
    constexpr int TB = 192;  // B-frag image of T = R @ K_inv

    const int b = blockIdx.x;
    const int l = threadIdx.x;
    const int m = l & 15;
    const int koff = (l & 16) * 2;   // 0 (k=0,1) or 32 (k=2,3)

    // Zero the whole scratch: provides all k=3 / row>=3 padding.
#pragma unroll
    for (int i = 0; i < 8; ++i) smem[l + 32 * i] = 0.0f;

    // Euler XYZ -> R = Rx @ Ry @ Rz (uniform; computed in registers)
    const float ea = pose[b * 6 + 0];
    const float eb = pose[b * 6 + 1];
    const float ec = pose[b * 6 + 2];
    const float ca = __cosf(ea), sa = __sinf(ea);
    const float cb = __cosf(eb), sb = __sinf(eb);
    const float cc = __cosf(ec), sc = __sinf(ec);
    const float R00 = cb * cc,                R01 = -cb * sc,               R02 = sb;
    const float R10 = ca * sc + sa * sb * cc, R11 = ca * cc - sa * sb * sc, R12 = -sa * cb;
    const float R20 = sa * sc - ca * sb * cc, R21 = sa * cc + ca * sb * sc, R22 = ca * cb;

    __syncthreads();

    if (l == 0) {
        // R as A-frag image: smem[RA + k*16 + row] = R[row][k]
        smem[RA + 0 * 16 + 0] = R00; smem[RA + 1 * 16 + 0] = R01; smem[RA + 2 * 16 + 0] = R02;
        smem[RA + 0 * 16 + 1] = R10; smem[RA + 1 * 16 + 1] = R11; smem[RA + 2 * 16 + 1] = R12;
        smem[RA + 0 * 16 + 2] = R20; smem[RA + 1 * 16 + 2] = R21; smem[RA + 2 * 16 + 2] = R22;
#pragma unroll
        for (int r = 0; r < 3; ++r)
#pragma unroll
            for (int c = 0; c < 3; ++c) {
                smem[KI + r * 16 + c] = Ki[b * 9 + r * 3 + c];  // B-frag: [k=r][n=c]
                smem[KA + c * 16 + r] = Km[b * 9 + r * 3 + c];  // A-frag: [k=c][m=r]
            }
    }
    __syncthreads();

    // ---- WMMA #1: T = R @ K_inv ----
    v2f aR, bKi;
    aR[0]  = smem[RA + koff + m];       aR[1]  = smem[RA + koff + 16 + m];
    bKi[0] = smem[KI + koff + m];       bKi[1] = smem[KI + koff + 16 + m];
    v8f zero8 = {};
    v8f T = __builtin_amdgcn_wmma_f32_16x16x4_f32(
        false, aR, false, bKi, (short)0, zero8, false, false);

    // T rows 0..2 (VGPRs T[0..2], lanes 0-15) -> B-frag image (row 3 stays 0)
    if (l < 16) {
        smem[TB + 0 * 16 + l] = T[0];
        smem[TB + 1 * 16 + l] = T[1];
        smem[TB + 2 * 16 + l] = T[2];
    }
    __syncthreads();

    // ---- WMMA #2: A = K @ T ----
    v2f aK, bT;
    aK[0] = smem[KA + koff + m];        aK[1] = smem[KA + koff + 16 + m];
    bT[0] = smem[TB + koff + m];        bT[1] = smem[TB + koff + 16 + m];
    v8f Af = __builtin_amdgcn_wmma_f32_16x16x4_f32(
        false, aK, false, bT, (short)0, zero8, false, false);

    // D[i][j] in VGPR i at lane j: lane j (<3) writes column j of rows 0..2.
    if (l < 3) {
        Aout[b * 12 + 0 + l] = Af[0];
        Aout[b * 12 + 3 + l] = Af[1];
        Aout[b * 12 + 6 + l] = Af[2];
    }
}

// ---------------------------------------------------------------------------
// Kernel 2: homography projection + bilinear gather. Bandwidth bound
// (~90 MB moved; ~4 us at 23.3 TB/s HBM; working set fits in 192 MB L2).
// One lane per output pixel; tap indices/weights computed once, reused for
// all 3 channels; output streamed with non-temporal stores.
// ---------------------------------------------------------------------------
__global__ __launch_bounds__(256) void project_kernel(
    const float* __restrict__ img, const float* __restrict__ z,
    const float* __restrict__ Amat, float* __restrict__ out)
{
    const int b = blockIdx.y;
    const int n = blockIdx.x * 256 + threadIdx.x;   // pixel index, row-major (y=n/W, x=n%W)

    // Uniform per block -> scalar loads
    const float* A = Amat + b * 12;
    const float A00 = A[0], A01 = A[1], A02 = A[2];
    const float A10 = A[3], A11 = A[4], A12 = A[5];
    const float A20 = A[6], A21 = A[7], A22 = A[8];

    const float zz = z[b * HW + n];

    // Reference quirk: source coords use xs = n // H, ys = n % H
    const int   q  = n / HH;
    const float xx = (float)q;
    const float yy = (float)(n - q * HH);

    const float u  = fmaf(A00, xx, fmaf(A01, yy, A02));
    const float v  = fmaf(A10, xx, fmaf(A11, yy, A12));
    const float wv = fmaf(A20, xx, fmaf(A21, yy, A22));
    const float px = zz * u, py = zz * v, pz = zz * wv;
    const float den = pz + EPSF;
    const float gxn = px / den;
    const float gyn = py / den;

    // Replicate the reference's normalize / denormalize round-trip exactly
    const float gridx = 2.0f * gxn / (float)WW - 1.0f;
    const float gridy = 2.0f * gyn / (float)HH - 1.0f;
    const float gx = ((gridx + 1.0f) * (float)WW - 1.0f) * 0.5f;
    const float gy = ((gridy + 1.0f) * (float)HH - 1.0f) * 0.5f;

    const float x0f = floorf(gx), y0f = floorf(gy);
    const float x1f = x0f + 1.0f, y1f = y0f + 1.0f;
    const float wx1 = gx - x0f, wx0 = 1.0f - wx1;
    const float wy1 = gy - y0f, wy0 = 1.0f - wy1;

    auto tap = [&](float xf, float yf, float wgt, int& idx, float& wout) {
        const bool valid = (xf >= 0.0f) && (xf < (float)WW) &&
                           (yf >= 0.0f) && (yf < (float)HH);
        const float xc = fminf(fmaxf(xf, 0.0f), (float)(WW - 1));
        const float yc = fminf(fmaxf(yf, 0.0f), (float)(HH - 1));
        idx  = (int)yc * WW + (int)xc;
        wout = valid ? wgt : 0.0f;
    };

    int   i00, i10, i01, i11;
    float w00, w10, w01, w11;
    tap(x0f, y0f, wx0 * wy0, i00, w00);
    tap(x1f, y0f, wx1 * wy0, i10, w10);
    tap(x0f, y1f, wx0 * wy1, i01, w01);
    tap(x1f, y1f, wx1 * wy1, i11, w11);

    const float* imgb = img + (size_t)b * (CHN * HW);
    float* outb = out + (size_t)b * (CHN * HW) + n;
#pragma unroll
    for (int c = 0; c < CHN; ++c) {
        const float* p = imgb + c * HW;
        const float r = p[i00] * w00 + p[i10] * w10 + p[i01] * w01 + p[i11] * w11;
        __builtin_nontemporal_store(r, outb + c * HW);
    }
}

extern "C" void kernel_launch(void* const* d_in, const int* in_sizes, int n_in,
                              void* d_out, int out_size, void* d_ws, size_t ws_size,
                              hipStream_t stream)
{
    (void)in_sizes; (void)n_in; (void)out_size; (void)ws_size;
    const float* img  = (const float*)d_in[0];   // (16,3,384,512)
    const float* z    = (const float*)d_in[1];   // (16,1,384,512)
    const float* pose = (const float*)d_in[2];   // (16,6)
    const float* K    = (const float*)d_in[3];   // (16,1,3,3)
    const float* Kinv = (const float*)d_in[4];   // (16,1,3,3)
    float* out = (float*)d_out;
    float* Aws = (float*)d_ws;                   // 16 * 12 floats

    compose_A_kernel<<<NB, 32, 0, stream>>>(pose, K, Kinv, Aws);

    dim3 grid(HW / 256, NB);
    project_kernel<<<grid, 256, 0, stream>>>(img, z, Aws, out);
}